// FiniteDifference_23957327577310
// MI455X (gfx1250) — compile-verified
//
#include <hip/hip_runtime.h>

// FiniteDifference on [B,T,H,W,C] = [4,16,128,128,16] fp32.
// Memory-bound: 64 MiB in + 192 MiB out => ~11.5 us floor at 23.3 TB/s.
// Strategy: one block per (b,t,h) row; async-stage the row into LDS (CDNA5
// global_load_async_to_lds_b128), w-diffs from LDS, h/t-diffs from global
// (input fits in 192 MB L2 so cross-row re-reads are L2 hits), non-temporal
// b128 stores for the pure-streaming outputs.

typedef float f4 __attribute__((ext_vector_type(4)));
typedef int   v4i __attribute__((ext_vector_type(4)));

typedef __attribute__((address_space(1))) v4i* gv4i_ptr;  // global
typedef __attribute__((address_space(3))) v4i* lv4i_ptr;  // LDS

#ifndef __has_builtin
#define __has_builtin(x) 0
#endif

#if __has_builtin(__builtin_amdgcn_global_load_async_to_lds_b128)
#define HAS_ASYNC_LDS 1
#else
#define HAS_ASYNC_LDS 0
#endif

constexpr int Bn = 4, Tn = 16, Hn = 128, Wn = 128, Cn = 16;
constexpr int ROWF  = Wn * Cn;   // 2048 floats per (b,t,h) row
constexpr int ROWF4 = ROWF / 4;  // 512 float4 per row
constexpr int NTHREADS = 256;    // 8 wave32s

__global__ __launch_bounds__(NTHREADS)
void FiniteDifference_23957327577310_kernel(const float* __restrict__ x,
                                            float* __restrict__ out_dy,
                                            float* __restrict__ out_dx,
                                            float* __restrict__ out_dz) {
  __shared__ float row[ROWF];  // 8 KB: center row, serves both w-neighbors

  const int bid = blockIdx.x;            // 0 .. B*T*H-1
  const int h = bid % Hn;
  const int t = (bid / Hn) % Tn;
  const int b = bid / (Hn * Tn);

  const size_t rowStride = (size_t)ROWF;          // stride between h rows
  const size_t tStride   = rowStride * Hn;        // stride between t slices
  const size_t bStride   = tStride * Tn;
  const size_t base = (size_t)b * bStride + (size_t)t * tStride
                    + (size_t)h * rowStride;
  const float* __restrict__ xrow = x + base;

  // ---- Stage center row into LDS via CDNA5 async global->LDS b128 copies ----
  for (int i = threadIdx.x; i < ROWF4; i += NTHREADS) {
#if HAS_ASYNC_LDS
    __builtin_amdgcn_global_load_async_to_lds_b128(
        (gv4i_ptr)(xrow + i * 4),
        (lv4i_ptr)(&row[i * 4]),
        /*imm offset=*/0, /*cpol=*/0);
#else
    *(f4*)(&row[i * 4]) = *(const f4*)(xrow + i * 4);
#endif
  }
#if HAS_ASYNC_LDS
  __builtin_amdgcn_s_wait_asynccnt(0);   // my wave's async copies landed in LDS
#endif
  __syncthreads();                        // everyone's copies visible

  const f4 zero = {0.f, 0.f, 0.f, 0.f};
  const bool hm_ok = (h > 0), hp_ok = (h < Hn - 1);
  const bool tm_ok = (t > 0), tp_ok = (t < Tn - 1);

  for (int i = threadIdx.x; i < ROWF4; i += NTHREADS) {
    const int e = i * 4;        // float offset within the row
    const int w = e >> 4;       // 16 channels (4 quads) per w position

    // W-direction neighbors from LDS (ds_load_b128); zero 'same' padding.
    f4 wl = (w > 0)      ? *(const f4*)(&row[e - Cn]) : zero;
    f4 wr = (w < Wn - 1) ? *(const f4*)(&row[e + Cn]) : zero;

    // H-direction neighbors from global (adjacent rows -> L2 hits).
    f4 hm = hm_ok ? *(const f4*)(xrow - rowStride + e) : zero;
    f4 hp = hp_ok ? *(const f4*)(xrow + rowStride + e) : zero;

    // T-direction neighbors from global.
    f4 tm = tm_ok ? *(const f4*)(xrow - tStride + e) : zero;
    f4 tp = tp_ok ? *(const f4*)(xrow + tStride + e) : zero;

    const f4 vdy = hp - hm;     // filt = (-1,0,1): out[i] = x[i+1] - x[i-1]
    const f4 vdx = wr - wl;
    const f4 vdz = tp - tm;

    // Streaming outputs: non-temporal so the 192 MiB of writes don't evict
    // the (fully L2-resident) 64 MiB input.
    __builtin_nontemporal_store(vdy, (f4*)(out_dy + base + e));
    __builtin_nontemporal_store(vdx, (f4*)(out_dx + base + e));
    __builtin_nontemporal_store(vdz, (f4*)(out_dz + base + e));
  }
}

extern "C" void kernel_launch(void* const* d_in, const int* in_sizes, int n_in,
                              void* d_out, int out_size, void* d_ws, size_t ws_size,
                              hipStream_t stream) {
  const float* x = (const float*)d_in[0];
  float* out = (float*)d_out;
  const size_t N = (size_t)Bn * Tn * Hn * Wn * Cn;  // 16,777,216 per output

  // Tuple return order: (dy, dx, dz) concatenated flat in d_out.
  float* dy = out;
  float* dx = out + N;
  float* dz = out + 2 * N;

  const int nblocks = Bn * Tn * Hn;  // 8192 rows, one block each
  FiniteDifference_23957327577310_kernel<<<nblocks, NTHREADS, 0, stream>>>(
      x, dy, dx, dz);
}